// CTC_80118319940190
// MI455X (gfx1250) — compile-verified
//
#include <hip/hip_runtime.h>

#define MAXS 513
#define NEG_INF (-1e30f)

typedef __attribute__((ext_vector_type(2))) float v2f;
typedef __attribute__((ext_vector_type(8))) float v8f;

__device__ __forceinline__ float lse3(float a, float b, float c) {
  float m = fmaxf(a, fmaxf(b, c));
  return m + __logf(__expf(a - m) + __expf(b - m) + __expf(c - m));
}

// ---------------- Phase 1: forward alphas, store all T rows + logZ ----------
__global__ __launch_bounds__(512)
void ctc_fwd(const float* __restrict__ lp, const int* __restrict__ tgt,
             const int* __restrict__ tlen, float* __restrict__ alpha,
             float* __restrict__ logZ, int T, int B, int V, int L) {
  const int b = blockIdx.x;
  const int S = 2 * L + 1;
  if (S > MAXS) return;

  __shared__ float bufA[MAXS], bufB[MAXS];
  __shared__ int tps[MAXS];
  __shared__ unsigned char skip[MAXS];

  for (int s = threadIdx.x; s < S; s += blockDim.x)
    tps[s] = (s & 1) ? tgt[b * L + (s >> 1)] : 0;   // BLANK = 0 on evens
  __syncthreads();
  for (int s = threadIdx.x; s < S; s += blockDim.x) {
    int ok = 0;
    if (s >= 2) { int l = tps[s]; ok = (l != 0) && (l != tps[s - 2]); }
    skip[s] = (unsigned char)ok;
  }

  float* arow = alpha + (size_t)b * T * S;
  const float* lp0 = lp + (size_t)b * V;            // t = 0 row
  for (int s = threadIdx.x; s < S; s += blockDim.x) {
    float v = (s < 2) ? lp0[tps[s]] : NEG_INF;
    bufA[s] = v;
    arow[s] = v;
  }
  __syncthreads();

  float* cur = bufA;
  float* nxt = bufB;
  for (int t = 1; t < T; ++t) {
    const float* lpt = lp + ((size_t)t * B + b) * V;
    arow += S;
    for (int s = threadIdx.x; s < S; s += blockDim.x) {
      float a0 = cur[s];
      float a1 = (s >= 1) ? cur[s - 1] : NEG_INF;
      float a2 = (s >= 2 && skip[s]) ? cur[s - 2] : NEG_INF;
      float v = lse3(a0, a1, a2) + lpt[tps[s]];
      nxt[s] = v;
      arow[s] = v;
    }
    __syncthreads();                                 // double buffer: 1 barrier/step
    float* tmp = cur; cur = nxt; nxt = tmp;
  }

  if (threadIdx.x == 0) {
    int se = 2 * tlen[b];
    float a = cur[se], c = cur[se - 1];
    float m = fmaxf(a, c);
    logZ[b] = m + __logf(__expf(a - m) + __expf(c - m));
  }
}

// ------ Phase 2: backward betas fused with posterior-weighted loss ----------
__global__ __launch_bounds__(512)
void ctc_bwd(const float* __restrict__ lp, const int* __restrict__ tgt,
             const int* __restrict__ tlen, const float* __restrict__ alpha,
             const float* __restrict__ logZ, float* __restrict__ loss_b,
             int T, int B, int V, int L) {
  const int b = blockIdx.x;
  const int S = 2 * L + 1;
  if (S > MAXS) return;

  __shared__ float beta[MAXS], fbuf[MAXS];
  __shared__ int tps[MAXS];
  __shared__ unsigned char skip[MAXS];
  __shared__ float part[512];
  __shared__ float wred[32];

  for (int s = threadIdx.x; s < S; s += blockDim.x)
    tps[s] = (s & 1) ? tgt[b * L + (s >> 1)] : 0;
  __syncthreads();
  for (int s = threadIdx.x; s < S; s += blockDim.x) {
    int ok = 0;
    if (s >= 2) { int l = tps[s]; ok = (l != 0) && (l != tps[s - 2]); }
    skip[s] = (unsigned char)ok;
  }

  const int se = 2 * tlen[b];
  const float lz = logZ[b];
  for (int s = threadIdx.x; s < S; s += blockDim.x)
    beta[s] = (s == se || s == se - 1) ? 0.f : NEG_INF;
  __syncthreads();

  float acc = 0.f;
  for (int t = T - 1; t >= 0; --t) {
    const float* lpt  = lp + ((size_t)t * B + b) * V;
    const float* arow = alpha + ((size_t)b * T + t) * S;
    for (int s = threadIdx.x; s < S; s += blockDim.x) {
      float e  = lpt[tps[s]];                    // L2-resident gather
      float bt = beta[s];
      fbuf[s]  = bt + e;                         // f = beta_t + emit_t
      acc += __expf(arow[s] + bt - lz) * e;      // posterior * emit
    }
    if (t > 0)
      __builtin_prefetch(arow - S + (int)threadIdx.x, 0, 0);  // next alpha row
    __syncthreads();
    if (t > 0) {
      for (int s = threadIdx.x; s < S; s += blockDim.x) {
        float f0 = fbuf[s];
        float f1 = (s + 1 < S) ? fbuf[s + 1] : NEG_INF;
        float f2 = (s + 2 < S && skip[s + 2]) ? fbuf[s + 2] : NEG_INF;
        beta[s] = lse3(f0, f1, f2);
      }
      __syncthreads();
    }
  }

  // ---- block reduction of 512 partials via V_WMMA_F32_16X16X4_F32 ----
  // D = ones(16x4) * B(4x16) + C  =>  sum(all D entries) = 16 * sum(B entries),
  // which is invariant to the exact A/B VGPR layout.
  part[threadIdx.x] = acc;
  __syncthreads();
  if (threadIdx.x < 32) {                        // wave 0, EXEC all ones
    v2f aones = {1.f, 1.f};
    v8f c = {};
#pragma unroll
    for (int k = 0; k < 8; ++k) {
      v2f bm;
      bm.x = part[k * 64 + threadIdx.x];
      bm.y = part[k * 64 + 32 + threadIdx.x];
      c = __builtin_amdgcn_wmma_f32_16x16x4_f32(
          /*neg_a=*/false, aones, /*neg_b=*/false, bm,
          /*c_mod=*/(short)0, c, /*reuse_a=*/false, /*reuse_b=*/false);
    }
    wred[threadIdx.x] = c[0] + c[1] + c[2] + c[3] + c[4] + c[5] + c[6] + c[7];
  }
  __syncthreads();
  if (threadIdx.x == 0) {
    float tot = 0.f;
    for (int i = 0; i < 32; ++i) tot += wred[i];
    loss_b[b] = tot * (1.0f / 16.0f);
  }
}

// ---------------- Phase 3: deterministic final reduce -----------------------
__global__ void ctc_fin(const float* __restrict__ loss_b, float* __restrict__ out,
                        int B) {
  if (threadIdx.x == 0 && blockIdx.x == 0) {
    float s = 0.f;
    for (int i = 0; i < B; ++i) s += loss_b[i];
    out[0] = s * (-0.25f / (float)B);
  }
}

extern "C" void kernel_launch(void* const* d_in, const int* in_sizes, int n_in,
                              void* d_out, int out_size, void* d_ws, size_t ws_size,
                              hipStream_t stream) {
  const float* lp   = (const float*)d_in[0];   // [T,B,V] log-probs
  const int*   tgt  = (const int*)d_in[1];     // [B,L]
  const int*   tlen = (const int*)d_in[3];     // [B]
  // d_in[2] = input_lengths (unused by reference), d_in[4] = forget_rate (unused)

  const int V = 128;
  const int B = in_sizes[3];
  const int L = in_sizes[1] / B;
  const int T = in_sizes[0] / (B * V);
  const int S = 2 * L + 1;

  float* alpha = (float*)d_ws;                 // T*S*B floats (~269 MB)
  float* logZ  = alpha + (size_t)T * S * B;    // B floats
  float* lossb = logZ + B;                     // B floats

  ctc_fwd<<<B, 512, 0, stream>>>(lp, tgt, tlen, alpha, logZ, T, B, V, L);
  ctc_bwd<<<B, 512, 0, stream>>>(lp, tgt, tlen, alpha, logZ, lossb, T, B, V, L);
  ctc_fin<<<1, 64, 0, stream>>>(lossb, (float*)d_out, B);
}